// RelGraphConvLayer_79104707658002
// MI455X (gfx1250) — compile-verified
//
#include <hip/hip_runtime.h>

typedef float v2f __attribute__((ext_vector_type(2)));
typedef float v8f __attribute__((ext_vector_type(8)));

#define R_ 4
#define NB_ 2
#define D_ 128   // DIN == DOUT

// ---- W[r] = sum_b coeff[r,b]*basis[b]  (slots 0..3), slot 4 = loop_w copy
__global__ void build_W(const float* __restrict__ basis,
                        const float* __restrict__ coeff,
                        const float* __restrict__ loop_w,
                        float* __restrict__ Wall) {
    int i = blockIdx.x * blockDim.x + threadIdx.x;
    const int M = D_ * D_;
    if (i < R_ * M) {
        int r = i / M, j = i - r * M;
        float acc = 0.f;
#pragma unroll
        for (int b = 0; b < NB_; ++b)
            acc += coeff[r * NB_ + b] * basis[b * M + j];
        Wall[i] = acc;
    } else if (i < (R_ + 1) * M) {
        Wall[i] = loop_w[i - R_ * M];
    }
}

__global__ void zero_buf(float* __restrict__ p, int n) {
    int i = blockIdx.x * blockDim.x + threadIdx.x;
    if (i < n) p[i] = 0.f;
}

__global__ void deg_count(const int* __restrict__ dst, float* __restrict__ deg,
                          int E, int N) {
    int i = blockIdx.x * blockDim.x + threadIdx.x;
    if (i < R_ * E) {
        int r = i / E;
        atomicAdd(&deg[(size_t)r * N + dst[i]], 1.0f);
    }
}

__global__ void inv_deg(float* __restrict__ deg, int n) {
    int i = blockIdx.x * blockDim.x + threadIdx.x;
    if (i < n) deg[i] = 1.0f / fmaxf(deg[i], 1.0f);
}

// ---- y[m,:] = x[m,:] @ W (+ bias)
// 256 threads = 8 waves; W staged in LDS (pair-interleaved, bank-swizzled);
// each wave computes a 16x128 output stripe via 256 fp32 WMMAs.
__global__ __launch_bounds__(256) void wmma_gemm(
    const float* __restrict__ x,
    const float* __restrict__ W,      // [128,128] row-major
    const float* __restrict__ bias,   // read iff add_bias
    float* __restrict__ y,            // [N,128]
    int add_bias, int N)
{
    // W[k][c] stored at word (k>>1)*256 + (((c + (k>>1)*16)&127)<<1) + (k&1)
    // -> (W[ka][col], W[ka+1][col]) is one aligned ds_load_b64, conflict-free.
    __shared__ float Wl[D_ * D_];     // 64 KB

    for (int i = threadIdx.x; i < (D_ * D_) / 4; i += 256) {
        int k  = i >> 5;              // row 0..127
        int c0 = (i & 31) << 2;       // col 0..124 step 4
        float4 w4 = *reinterpret_cast<const float4*>(W + (size_t)k * D_ + c0);
        int pair = k >> 1;
        int lo   = k & 1;
        int base = pair << 8;
        int sw   = pair << 4;
        float wv[4] = {w4.x, w4.y, w4.z, w4.w};
#pragma unroll
        for (int j = 0; j < 4; ++j) {
            int col = c0 + j;
            Wl[base + (((col + sw) & 127) << 1) + lo] = wv[j];
        }
    }
    __syncthreads();

    int wave = threadIdx.x >> 5;
    int lane = threadIdx.x & 31;
    int mt = blockIdx.x * 8 + wave;
    int n_mtiles = (N + 15) >> 4;
    if (mt < n_mtiles) {
        int hi  = lane >> 4;          // 0: K={0,1}, 1: K={2,3} in each K-quad
        int lan = lane & 15;
        int m = mt * 16 + lan;
        int mclamp = m < N ? m : N - 1;
        const float* xrow = x + (size_t)mclamp * D_;

        v8f c[8];
#pragma unroll
        for (int n = 0; n < 8; ++n) c[n] = (v8f){0,0,0,0,0,0,0,0};

        for (int kb = 0; kb < D_; kb += 4) {
            int ka = kb + (hi << 1);
            v2f a = *reinterpret_cast<const v2f*>(xrow + ka);   // x[m][ka..ka+1]
            int pair = ka >> 1;
            int base = pair << 8;
            int sw   = pair << 4;
#pragma unroll
            for (int n = 0; n < 8; ++n) {
                int col = n * 16 + lan;
                v2f b = *reinterpret_cast<const v2f*>(
                    &Wl[base + (((col + sw) & 127) << 1)]);
                c[n] = __builtin_amdgcn_wmma_f32_16x16x4_f32(
                    false, a, false, b, (short)0, c[n], false, false);
            }
        }
#pragma unroll
        for (int n = 0; n < 8; ++n) {
            int col = n * 16 + lan;
            float bv = add_bias ? bias[col] : 0.f;
#pragma unroll
            for (int v = 0; v < 8; ++v) {
                int row = mt * 16 + v + (hi << 3);
                if (row < N) y[(size_t)row * D_ + col] = c[n][v] + bv;
            }
        }
    }
}

// ---- one wave per edge: out[dst] += Y[src] * invdeg[dst]  (float4 per lane)
__global__ __launch_bounds__(256) void scatter_edges(
    const int* __restrict__ src, const int* __restrict__ dst,
    const float* __restrict__ Y, const float* __restrict__ invdeg,
    float* __restrict__ out, int E)
{
    int gw   = (int)((blockIdx.x * 256u + threadIdx.x) >> 5);
    int lane = threadIdx.x & 31;
    if (gw >= E) return;
    int s = src[gw];
    int d = dst[gw];
    float sc = invdeg[d];
    const float4* yv = reinterpret_cast<const float4*>(Y + (size_t)s * D_);
    float4 v = yv[lane];
    float* o = out + (size_t)d * D_ + lane * 4;
    atomicAdd(o + 0, v.x * sc);
    atomicAdd(o + 1, v.y * sc);
    atomicAdd(o + 2, v.z * sc);
    atomicAdd(o + 3, v.w * sc);
}

__global__ void relu_k(float* __restrict__ p, int n) {
    int i = blockIdx.x * blockDim.x + threadIdx.x;
    if (i < n) p[i] = fmaxf(p[i], 0.f);
}

extern "C" void kernel_launch(void* const* d_in, const int* in_sizes, int n_in,
                              void* d_out, int out_size, void* d_ws, size_t ws_size,
                              hipStream_t stream) {
    (void)n_in; (void)out_size; (void)ws_size;
    const float* x        = (const float*)d_in[0];
    const int*   edge_src = (const int*)d_in[1];
    const int*   edge_dst = (const int*)d_in[2];
    const float* basis    = (const float*)d_in[3];
    const float* coeff    = (const float*)d_in[4];
    const float* loop_w   = (const float*)d_in[5];
    const float* loop_b   = (const float*)d_in[6];

    const int N = in_sizes[0] / D_;
    const int E = in_sizes[1] / R_;

    // workspace layout (floats): Wall[(R+1)*128*128] | deg[R*N] | Y[N*128]
    float* Wall = (float*)d_ws;
    float* deg  = Wall + (size_t)(R_ + 1) * D_ * D_;
    float* Y    = deg  + (size_t)R_ * N;

    const int T = 256;
    build_W<<<((R_ + 1) * D_ * D_ + T - 1) / T, T, 0, stream>>>(basis, coeff, loop_w, Wall);
    zero_buf<<<(R_ * N + T - 1) / T, T, 0, stream>>>(deg, R_ * N);
    deg_count<<<(R_ * E + T - 1) / T, T, 0, stream>>>(edge_dst, deg, E, N);
    inv_deg<<<(R_ * N + T - 1) / T, T, 0, stream>>>(deg, R_ * N);

    const int n_mtiles = (N + 15) / 16;
    const int gemm_blocks = (n_mtiles + 7) / 8;

    // out = x @ loop_w + bias   (initializes accumulator)
    wmma_gemm<<<gemm_blocks, 256, 0, stream>>>(
        x, Wall + (size_t)R_ * D_ * D_, loop_b, (float*)d_out, 1, N);

    const int sc_blocks = (E * 32 + 255) / 256;
    for (int r = 0; r < R_; ++r) {
        wmma_gemm<<<gemm_blocks, 256, 0, stream>>>(
            x, Wall + (size_t)r * D_ * D_, loop_b, Y, 0, N);
        scatter_edges<<<sc_blocks, 256, 0, stream>>>(
            edge_src + (size_t)r * E, edge_dst + (size_t)r * E,
            Y, deg + (size_t)r * N, (float*)d_out, E);
    }
    relu_k<<<(N * D_ + T - 1) / T, T, 0, stream>>>((float*)d_out, N * D_);
}